// GCN1_7739531067711
// MI455X (gfx1250) — compile-verified
//
#include <hip/hip_runtime.h>
#include <math.h>

typedef __attribute__((ext_vector_type(2))) float v2f;
typedef __attribute__((ext_vector_type(8))) float v8f;

#define N_NODES_C 100000
#define N_EDGES_C 1000000
#define N_PAIRS_C 200000
#define DDIM 64
#define WSTRIDE 80  // v2f row stride for K-paired LDS weights (bank-conflict-free)

static __device__ __forceinline__ v8f wmma4(v2f a, v2f b, v8f c) {
  // D = A(16x4,f32) x B(4x16,f32) + C(16x16,f32)
  return __builtin_amdgcn_wmma_f32_16x16x4_f32(false, a, false, b, (short)0, c,
                                               false, false);
}

// ---------------- degree accumulation + rsqrt ----------------
__global__ __launch_bounds__(256) void deg_kernel(const int* __restrict__ src,
                                                  const int* __restrict__ dst,
                                                  float* __restrict__ outdeg,
                                                  float* __restrict__ indeg, int nE) {
  int e = blockIdx.x * blockDim.x + threadIdx.x;
  if (e < nE) {
    atomicAdd(&outdeg[src[e]], 1.0f);
    atomicAdd(&indeg[dst[e]], 1.0f);
  }
}

__global__ __launch_bounds__(256) void rsqrt_kernel(float* __restrict__ a,
                                                    float* __restrict__ b, int n) {
  int i = blockIdx.x * blockDim.x + threadIdx.x;
  if (i < n) {
    a[i] = rsqrtf(fmaxf(a[i], 1.0f));
    b[i] = rsqrtf(fmaxf(b[i], 1.0f));
  }
}

// ---------------- T[N,64] = (H * scale[:,None]) @ W[64,64]  (fp32 WMMA) ----------------
__global__ __launch_bounds__(256) void gemm64_kernel(const float* __restrict__ H,
                                                     const float* __restrict__ scale,
                                                     const float* __restrict__ W,
                                                     float* __restrict__ T, int nRows) {
  // W staged as K-pairs: sW[k2*WSTRIDE + col] = { W[2k2][col], W[2k2+1][col] }
  __shared__ v2f sW[32 * WSTRIDE];  // 20 KB
  for (int idx = threadIdx.x; idx < 32 * DDIM; idx += 256) {
    int k2 = idx >> 6, col = idx & 63;
    v2f w;
    w.x = W[(2 * k2) * DDIM + col];
    w.y = W[(2 * k2 + 1) * DDIM + col];
    sW[k2 * WSTRIDE + col] = w;
  }
  __syncthreads();

  const int wave = threadIdx.x >> 5;
  const int lane = threadIdx.x & 31;
  const int m = lane & 15;
  const int hi = lane >> 4;  // 0 or 1

  int rowBase = (blockIdx.x * 8 + wave) * 16;
  if (rowBase >= nRows) return;  // uniform per wave; after the barrier

  int row = rowBase + m;
  bool rv = row < nRows;
  int rc = rv ? row : nRows - 1;
  float sc = rv ? scale[rc] : 0.0f;
  const float* hr = H + (size_t)rc * DDIM;

  v8f acc[4];
#pragma unroll
  for (int c = 0; c < 4; ++c) acc[c] = 0.0f;

#pragma unroll
  for (int k0 = 0; k0 < DDIM; k0 += 4) {
    int ka = k0 + 2 * hi;  // even
    v2f a = *(const v2f*)(hr + ka);
    a.x *= sc;
    a.y *= sc;
    const v2f* wrow = sW + (ka >> 1) * WSTRIDE + m;
#pragma unroll
    for (int c = 0; c < 4; ++c) {
      v2f b = wrow[c * 16];  // single 8B-aligned ds_load_b64
      acc[c] = wmma4(a, b, acc[c]);
    }
  }

#pragma unroll
  for (int r = 0; r < 8; ++r) {
    int orow = rowBase + r + 8 * hi;
    if (orow < nRows) {
      float* o = T + (size_t)orow * DDIM + m;
#pragma unroll
      for (int c = 0; c < 4; ++c) o[c * 16] = acc[c][r];
    }
  }
}

// ---------------- agg[dst] += T[src]  (one wave per edge) ----------------
__global__ __launch_bounds__(256) void scatter_kernel(const float* __restrict__ T,
                                                      const int* __restrict__ src,
                                                      const int* __restrict__ dst,
                                                      float* __restrict__ agg, int nE) {
  int gid = blockIdx.x * blockDim.x + threadIdx.x;
  int e = gid >> 5;
  int lane = gid & 31;
  if (e < nE) {
    int s = src[e], d = dst[e];
    const float* ts = T + (size_t)s * DDIM;
    float* ad = agg + (size_t)d * DDIM;
    atomicAdd(&ad[lane], ts[lane]);
    atomicAdd(&ad[lane + 32], ts[lane + 32]);
  }
}

// ---------------- Hout = act(agg * in_isqrt[:,None] + bias) ----------------
__global__ __launch_bounds__(256) void finalize_kernel(const float* __restrict__ agg,
                                                       const float* __restrict__ inI,
                                                       const float* __restrict__ bias,
                                                       float* __restrict__ Hout, int n,
                                                       int do_relu) {
  int idx = blockIdx.x * blockDim.x + threadIdx.x;
  if (idx < n * DDIM) {
    int i = idx >> 6, j = idx & 63;
    float v = agg[idx] * inI[i] + bias[j];
    if (do_relu) v = fmaxf(v, 0.0f);
    Hout[idx] = v;
  }
}

// ---------------- fused pair head: softmax(relu([h[g1],h[g2]]@Wfc1+b1)@Wfc2+b2) -------
__global__ __launch_bounds__(128) void fc_head_kernel(
    const float* __restrict__ h, const int* __restrict__ g1, const int* __restrict__ g2,
    const float* __restrict__ Wfc1, const float* __restrict__ bfc1,
    const float* __restrict__ Wfc2, const float* __restrict__ bfc2,
    float* __restrict__ out, int nPairs) {
  __shared__ v2f sW[64 * WSTRIDE];  // 40 KB, K-paired Wfc1
  __shared__ float sZ[4][16][68];   // ~17 KB, padded vs bank conflicts
  for (int idx = threadIdx.x; idx < 64 * DDIM; idx += 128) {
    int k2 = idx >> 6, col = idx & 63;
    v2f w;
    w.x = Wfc1[(2 * k2) * DDIM + col];
    w.y = Wfc1[(2 * k2 + 1) * DDIM + col];
    sW[k2 * WSTRIDE + col] = w;
  }

  const int wave = threadIdx.x >> 5;
  const int lane = threadIdx.x & 31;
  const int m = lane & 15;
  const int hi = lane >> 4;

  int nTiles = (nPairs + 15) / 16;
  int tile = blockIdx.x * 4 + wave;
  bool tv = tile < nTiles;
  int tc = tv ? tile : nTiles - 1;

  int pr = tc * 16 + m;
  int pc = pr < nPairs ? pr : nPairs - 1;
  const float* h1 = h + (size_t)g1[pc] * DDIM;
  const float* h2 = h + (size_t)g2[pc] * DDIM;
  __syncthreads();

  v8f acc[4];
#pragma unroll
  for (int c = 0; c < 4; ++c) acc[c] = 0.0f;

#pragma unroll
  for (int k0 = 0; k0 < 2 * DDIM; k0 += 4) {
    int ka = k0 + 2 * hi;  // even; ka,ka+1 in same half
    const float* hp = (ka < DDIM) ? h1 : h2;
    int kb = (ka < DDIM) ? ka : ka - DDIM;
    v2f a = *(const v2f*)(hp + kb);
    const v2f* wrow = sW + (ka >> 1) * WSTRIDE + m;
#pragma unroll
    for (int c = 0; c < 4; ++c) {
      v2f b = wrow[c * 16];
      acc[c] = wmma4(a, b, acc[c]);
    }
  }

  // z = relu(acc + bfc1) -> LDS (per-wave slice)
#pragma unroll
  for (int r = 0; r < 8; ++r) {
    int zr = r + 8 * hi;
#pragma unroll
    for (int c = 0; c < 4; ++c) {
      float z = acc[c][r] + bfc1[c * 16 + m];
      sZ[wave][zr][c * 16 + m] = fmaxf(z, 0.0f);
    }
  }
  __syncthreads();

  if (hi == 0) {
    float l0 = bfc2[0], l1 = bfc2[1];
#pragma unroll 8
    for (int j = 0; j < DDIM; ++j) {
      float z = sZ[wave][m][j];
      l0 += z * Wfc2[j * 2 + 0];
      l1 += z * Wfc2[j * 2 + 1];
    }
    float mx = fmaxf(l0, l1);
    float e0 = __expf(l0 - mx), e1 = __expf(l1 - mx);
    float inv = 1.0f / (e0 + e1);
    int prr = tile * 16 + m;
    if (tv && prr < nPairs) {
      out[prr * 2 + 0] = e0 * inv;
      out[prr * 2 + 1] = e1 * inv;
    }
  }
}

extern "C" void kernel_launch(void* const* d_in, const int* in_sizes, int n_in,
                              void* d_out, int out_size, void* d_ws, size_t ws_size,
                              hipStream_t stream) {
  const float* x    = (const float*)d_in[0];
  const int*   src  = (const int*)d_in[1];
  const int*   dst  = (const int*)d_in[2];
  const int*   g1   = (const int*)d_in[3];
  const int*   g2   = (const int*)d_in[4];
  const float* W1   = (const float*)d_in[5];
  const float* b1   = (const float*)d_in[6];
  const float* W2   = (const float*)d_in[7];
  const float* b2   = (const float*)d_in[8];
  const float* W3   = (const float*)d_in[9];
  const float* b3   = (const float*)d_in[10];
  const float* Wfc1 = (const float*)d_in[11];
  const float* bfc1 = (const float*)d_in[12];
  const float* Wfc2 = (const float*)d_in[13];
  const float* bfc2 = (const float*)d_in[14];
  float* out = (float*)d_out;

  const int N = N_NODES_C, E = N_EDGES_C, P = N_PAIRS_C;

  // workspace layout (floats)
  float* ws = (float*)d_ws;
  const size_t PAD_N = 102400;  // padded N for the two norm vectors
  float* outI = ws;                                   // [N]
  float* inI  = ws + PAD_N;                           // [N]
  float* bufT = ws + 2 * PAD_N;                       // [N,64]
  float* bufH = bufT + (size_t)N * DDIM;              // [N,64]
  float* bufA = bufH + (size_t)N * DDIM;              // [N,64]
  (void)ws_size; (void)in_sizes; (void)n_in; (void)out_size;

  const int gemmBlocks = (((N + 15) / 16) + 7) / 8;     // 8 waves/block, 16 rows/wave
  const int scatBlocks = (E * 32 + 255) / 256;          // 1 wave per edge
  const int finBlocks  = (N * DDIM + 255) / 256;
  const int headBlocks = (((P + 15) / 16) + 3) / 4;     // 4 waves/block

  // degrees -> rsqrt norms
  hipMemsetAsync(outI, 0, 2 * PAD_N * sizeof(float), stream);
  deg_kernel<<<(E + 255) / 256, 256, 0, stream>>>(src, dst, outI, inI, E);
  rsqrt_kernel<<<(N + 255) / 256, 256, 0, stream>>>(outI, inI, N);

  // ---- layer 1 ----
  gemm64_kernel<<<gemmBlocks, 256, 0, stream>>>(x, outI, W1, bufT, N);
  hipMemsetAsync(bufA, 0, (size_t)N * DDIM * sizeof(float), stream);
  scatter_kernel<<<scatBlocks, 256, 0, stream>>>(bufT, src, dst, bufA, E);
  finalize_kernel<<<finBlocks, 256, 0, stream>>>(bufA, inI, b1, bufH, N, 1);

  // ---- layer 2 ----
  gemm64_kernel<<<gemmBlocks, 256, 0, stream>>>(bufH, outI, W2, bufT, N);
  hipMemsetAsync(bufA, 0, (size_t)N * DDIM * sizeof(float), stream);
  scatter_kernel<<<scatBlocks, 256, 0, stream>>>(bufT, src, dst, bufA, E);
  finalize_kernel<<<finBlocks, 256, 0, stream>>>(bufA, inI, b2, bufH, N, 1);

  // ---- layer 3 (no relu) ----
  gemm64_kernel<<<gemmBlocks, 256, 0, stream>>>(bufH, outI, W3, bufT, N);
  hipMemsetAsync(bufA, 0, (size_t)N * DDIM * sizeof(float), stream);
  scatter_kernel<<<scatBlocks, 256, 0, stream>>>(bufT, src, dst, bufA, E);
  finalize_kernel<<<finBlocks, 256, 0, stream>>>(bufA, inI, b3, bufH, N, 0);

  // ---- fused pair head ----
  fc_head_kernel<<<headBlocks, 128, 0, stream>>>(bufH, g1, g2, Wfc1, bfc1, Wfc2,
                                                 bfc2, out, P);
}